// CrossEntropyLoss_57767310131345
// MI455X (gfx1250) — compile-verified
//
#include <hip/hip_runtime.h>
#include <stdint.h>
#include <math.h>

#define ROWS   4096
#define COLS   32000
#define T1     320                 // 10 wave32s per row-block
#define CHUNK  (T1 * 4)            // 1280 floats per async chunk
#define NCHUNK (COLS / CHUNK)      // 25 chunks, exact
#define DEPTH  4                   // async pipeline depth (20 KB LDS of 320 KB)

// ---- CDNA5 async global->LDS copy (ASYNCcnt path), inline asm per CDNA5 ISA ----
__device__ __forceinline__ void async_load_b128(uint32_t lds_off, const float* gptr) {
    // GV mode: 64-bit per-lane global address, VDST holds per-lane LDS byte address.
    asm volatile("global_load_async_to_lds_b128 %0, %1, off"
                 :: "v"(lds_off), "v"((uint64_t)(uintptr_t)gptr)
                 : "memory");
}

template<int N>
__device__ __forceinline__ void wait_asynccnt() {
    asm volatile("s_wait_asynccnt %0" :: "i"(N) : "memory");
}

// Online log-sum-exp accumulate: usually 1 exp per element (max updates are rare).
__device__ __forceinline__ void online(float v, float& m, float& s) {
    if (v <= m) {
        s += __expf(v - m);
    } else {
        s = s * __expf(m - v) + 1.0f;   // first element: s=0, __expf(-inf)=0 -> s=1
        m = v;
    }
}

__global__ __launch_bounds__(T1)
void ce_row_kernel(const float* __restrict__ logits,
                   const int*   __restrict__ target,
                   float*       __restrict__ row_loss) {
    __shared__ float4 stage[DEPTH][T1];
    __shared__ float  red_m[T1 / 32];
    __shared__ float  red_s[T1 / 32];

    const int tid = threadIdx.x;
    const int row = blockIdx.x;
    const float* rowp = logits + (size_t)row * COLS;

    // Prime the async pipeline: 4 chunks in flight per wave.
#pragma unroll
    for (int c = 0; c < DEPTH; ++c) {
        uint32_t off = (uint32_t)(uintptr_t)&stage[c][tid];
        async_load_b128(off, rowp + c * CHUNK + tid * 4);
    }

    float m = -INFINITY, s = 0.0f;

    for (int c = 0; c < NCHUNK; ++c) {
        // Chunk c is done when <= (#newer outstanding) asyncs remain for this wave.
        int rem = NCHUNK - 1 - c;
        if      (rem >= 3) wait_asynccnt<3>();
        else if (rem == 2) wait_asynccnt<2>();
        else if (rem == 1) wait_asynccnt<1>();
        else               wait_asynccnt<0>();

        float4 v = stage[c % DEPTH][tid];   // ds_load_b128, lane reads its own staged data
        online(v.x, m, s);
        online(v.y, m, s);
        online(v.z, m, s);
        online(v.w, m, s);

        int nxt = c + DEPTH;
        if (nxt < NCHUNK) {
            uint32_t off = (uint32_t)(uintptr_t)&stage[nxt % DEPTH][tid];
            async_load_b128(off, rowp + nxt * CHUNK + tid * 4);
        }
    }

    // Wave32 log-sum-exp reduction (all lanes have finite m after 100 elements).
#pragma unroll
    for (int d = 16; d >= 1; d >>= 1) {
        float om = __shfl_xor(m, d, 32);
        float os = __shfl_xor(s, d, 32);
        float mn = fmaxf(m, om);
        s = s * __expf(m - mn) + os * __expf(om - mn);
        m = mn;
    }

    if ((tid & 31) == 0) {
        red_m[tid >> 5] = m;
        red_s[tid >> 5] = s;
    }
    __syncthreads();

    if (tid == 0) {
        float M = red_m[0], S = red_s[0];
#pragma unroll
        for (int w = 1; w < T1 / 32; ++w) {
            float om = red_m[w], os = red_s[w];
            float mn = fmaxf(M, om);
            S = S * __expf(M - mn) + os * __expf(om - mn);
            M = mn;
        }
        int   t    = target[row];
        float xt   = rowp[t];
        float logZ = M + logf(S);
        float sc   = (t != 0) ? 1.5f : 1.0f;
        row_loss[row] = -(xt - logZ) * sc;
    }
}

__global__ __launch_bounds__(256)
void ce_mean_kernel(const float* __restrict__ row_loss, float* __restrict__ out) {
    __shared__ float sm[256];
    float acc = 0.0f;
    for (int i = threadIdx.x; i < ROWS; i += 256)
        acc += row_loss[i];
    sm[threadIdx.x] = acc;
    __syncthreads();
#pragma unroll
    for (int d = 128; d >= 1; d >>= 1) {
        if (threadIdx.x < d) sm[threadIdx.x] += sm[threadIdx.x + d];
        __syncthreads();
    }
    if (threadIdx.x == 0)
        out[0] = sm[0] * (1.0f / (float)ROWS);
}

extern "C" void kernel_launch(void* const* d_in, const int* in_sizes, int n_in,
                              void* d_out, int out_size, void* d_ws, size_t ws_size,
                              hipStream_t stream) {
    (void)in_sizes; (void)n_in; (void)out_size; (void)ws_size;
    const float* logits = (const float*)d_in[0];
    const int*   target = (const int*)d_in[1];
    float*       out    = (float*)d_out;
    float*       ws     = (float*)d_ws;   // ROWS floats of scratch (16 KB)

    ce_row_kernel<<<ROWS, T1, 0, stream>>>(logits, target, ws);
    ce_mean_kernel<<<1, 256, 0, stream>>>(ws, out);
}